// ConfPredModule_33397665694030
// MI455X (gfx1250) — compile-verified
//
#include <hip/hip_runtime.h>
#include <math.h>

typedef __attribute__((ext_vector_type(2))) float v2f;
typedef __attribute__((ext_vector_type(8))) float v8f;

#define D_FEAT 128
#define NBIN   65    // bins 0..63 = CDF start index, bin 64 = "twist beyond all thresholds"

// ---------------------------------------------------------------------------
// Kernel 1: one wave32 per edge.
//   dist  = ||x[dst]-x[src]||  (128-dim, float4 per lane, shfl_xor reduce)
//   dist2 = ||pos[dst]-pos[src]|| (3-dim, lane 0)
//   k0    = ceil(|dist-dist2| * 126)  clamped to [0,64]   (thr[k] = 0.5*k/63)
//   hist[dst][k0] += 1   (single u32 atomic instead of 64 float atomics)
// ---------------------------------------------------------------------------
__global__ __launch_bounds__(256)
void confpred_edge_hist(const float* __restrict__ x,
                        const float* __restrict__ pos,
                        const int*   __restrict__ edge_index,
                        unsigned int* __restrict__ hist,
                        int E)
{
    int e    = blockIdx.x * (blockDim.x >> 5) + (threadIdx.x >> 5);
    int lane = threadIdx.x & 31;
    if (e >= E) return;

    int src = edge_index[e];
    int dst = edge_index[E + e];

    const float4* xs = (const float4*)(x + (size_t)src * D_FEAT);
    const float4* xd = (const float4*)(x + (size_t)dst * D_FEAT);
    float4 a = xs[lane];      // wave reads 512 contiguous bytes (global_load_b128)
    float4 b = xd[lane];
    float d0 = b.x - a.x, d1 = b.y - a.y, d2 = b.z - a.z, d3 = b.w - a.w;
    float acc = d0*d0 + d1*d1 + d2*d2 + d3*d3;
#pragma unroll
    for (int off = 16; off > 0; off >>= 1)
        acc += __shfl_xor(acc, off, 32);

    if (lane == 0) {
        float p0 = pos[(size_t)dst*3+0] - pos[(size_t)src*3+0];
        float p1 = pos[(size_t)dst*3+1] - pos[(size_t)src*3+1];
        float p2 = pos[(size_t)dst*3+2] - pos[(size_t)src*3+2];
        float dist  = sqrtf(acc);
        float dist2 = sqrtf(p0*p0 + p1*p1 + p2*p2);
        float twist = fabsf(dist - dist2);
        // twist <= thr[k] = k*(0.5/63)  <=>  k >= twist*126
        int k0 = (int)ceilf(twist * 126.0f);
        if (k0 > 64) k0 = 64;
        if (k0 < 0)  k0 = 0;
        atomicAdd(&hist[(size_t)dst * NBIN + (unsigned)k0], 1u);
    }
}

// ---------------------------------------------------------------------------
// Kernel 2: one wave32 per 16-node tile.
//   lanes 0..15: prefix-sum 64 histogram bins -> sum_msg; cnt = run + bin64;
//                mean_msg = sum_msg / max(cnt,1); stage [16 x 128] msg in LDS
//   all lanes  : D(16x16) = msg(16x128) @ Wt(128x16) via 32x
//                V_WMMA_F32_16X16X4_F32. W staged in LDS with an extra
//                all-zero row so cols 2..15 read the zero row -> every step
//                is two unconditional ds_load_b64 + wmma (no EXEC toggling).
//   epilogue   : +bias, 2-way softmax, coalesced msg store
// ---------------------------------------------------------------------------
__global__ __launch_bounds__(32)
void confpred_node_wmma(const unsigned int* __restrict__ hist,
                        const float* __restrict__ W,     // [2][128] row-major
                        const float* __restrict__ bias,  // [2]
                        float* __restrict__ out_msg,     // [N][128]
                        float* __restrict__ out_conf,    // [N][2]
                        int N)
{
    __shared__ float lds_msg[16 * 128];
    __shared__ float lds_W[3 * 128];       // rows 0,1 = W; row 2 = zeros
    __shared__ float lds_logit[16 * 2];

    int lane  = threadIdx.x;       // 0..31
    int n0    = blockIdx.x * 16;
    int nrows = N - n0; if (nrows > 16) nrows = 16;

    // stage W (256 floats) + zero row (128 floats) into LDS
#pragma unroll
    for (int i = 0; i < 8; ++i)
        lds_W[lane + 32*i] = W[lane + 32*i];
#pragma unroll
    for (int i = 0; i < 4; ++i)
        lds_W[256 + lane + 32*i] = 0.0f;

    if (lane < 16 && lane < nrows) {
        const unsigned int* h = hist + (size_t)(n0 + lane) * NBIN;
        unsigned int run = 0;
        for (int k = 0; k < 64; ++k) {
            run += h[k];
            lds_msg[lane*128 + 64 + k] = (float)run;          // sum_msg (aggr='add')
        }
        unsigned int cnt = run + h[64];
        float inv = 1.0f / (float)(cnt > 0u ? cnt : 1u);
        for (int k = 0; k < 64; ++k)
            lds_msg[lane*128 + k] = lds_msg[lane*128 + 64 + k] * inv;  // mean_msg
    } else if (lane < 16) {
        for (int k = 0; k < 128; ++k) lds_msg[lane*128 + k] = 0.0f;   // pad tail tile
    }
    __syncthreads();

    // f32 WMMA A-layout (16x4): lanes 0-15 hold K=0,1 ; lanes 16-31 hold K=2,3
    int rc     = lane & 15;                 // A row / B col
    int khalf  = (lane >> 4) << 1;          // 0 or 2
    int wrow   = (rc < 2) ? rc : 2;         // cols >=2 read the zero row
    const float* arow = lds_msg + rc * 128 + khalf;
    const float* brow = lds_W  + wrow * 128 + khalf;
    v8f c = {};
#pragma unroll
    for (int s = 0; s < 32; ++s) {
        v2f av, bv;
        av[0] = arow[4*s];     av[1] = arow[4*s + 1];   // ds_load_b64
        bv[0] = brow[4*s];     bv[1] = brow[4*s + 1];   // ds_load_b64
        c = __builtin_amdgcn_wmma_f32_16x16x4_f32(false, av, false, bv,
                                                  (short)0, c, false, false);
    }

    // D layout: VGPR v, lanes 0-15 -> M=v, N=lane ; lanes 16-31 -> M=v+8
    if (rc < 2) {
        int mbase = (lane >> 4) << 3;       // 0 or 8
#pragma unroll
        for (int v = 0; v < 8; ++v)
            lds_logit[(mbase + v)*2 + rc] = c[v] + bias[rc];
    }
    __syncthreads();

    // coalesced msg store
    float* om = out_msg + (size_t)n0 * 128;
#pragma unroll
    for (int i = 0; i < 64; ++i) {
        int idx = lane + 32*i;
        if (idx < nrows * 128) om[idx] = lds_msg[idx];
    }

    // softmax over 2 classes
    if (lane < nrows) {
        float l0 = lds_logit[lane*2+0];
        float l1 = lds_logit[lane*2+1];
        float m  = fmaxf(l0, l1);
        float e0 = __expf(l0 - m);
        float e1 = __expf(l1 - m);
        float inv = 1.0f / (e0 + e1);
        out_conf[(size_t)(n0 + lane)*2 + 0] = e0 * inv;
        out_conf[(size_t)(n0 + lane)*2 + 1] = e1 * inv;
    }
}

// ---------------------------------------------------------------------------
extern "C" void kernel_launch(void* const* d_in, const int* in_sizes, int n_in,
                              void* d_out, int out_size, void* d_ws, size_t ws_size,
                              hipStream_t stream)
{
    const float* x   = (const float*)d_in[0];   // [N,128]
    const float* pos = (const float*)d_in[1];   // [N,3]
    const float* W   = (const float*)d_in[2];   // [2,128]
    const float* b   = (const float*)d_in[3];   // [2]
    // d_in[4] = batch (unused: single graph)
    const int* edge_index = (const int*)d_in[5];// [2,E]

    int N = in_sizes[0] / D_FEAT;
    int E = in_sizes[5] / 2;

    unsigned int* hist = (unsigned int*)d_ws;   // [N,65] u32 = 26 MB, L2-resident
    hipMemsetAsync(hist, 0, (size_t)N * NBIN * sizeof(unsigned int), stream);

    const int wavesPerBlock = 8;                // 256 threads
    int eblocks = (E + wavesPerBlock - 1) / wavesPerBlock;
    confpred_edge_hist<<<eblocks, 32 * wavesPerBlock, 0, stream>>>(
        x, pos, edge_index, hist, E);

    float* out_msg  = (float*)d_out;
    float* out_conf = out_msg + (size_t)N * 2 * 64;
    int nblocks = (N + 15) / 16;
    confpred_node_wmma<<<nblocks, 32, 0, stream>>>(hist, W, b, out_msg, out_conf, N);
}